// Encoder_Head_77343771066713
// MI455X (gfx1250) — compile-verified
//
#include <hip/hip_runtime.h>
#include <hip/hip_bf16.h>
#include <cstdint>

#define NPTS 2048
#define KNBR 40
// 1/sqrt(1 + 1e-5), the BN affine scale factor
#define BN_RSQ 0.9999950000374997f

typedef float v2f __attribute__((ext_vector_type(2)));
typedef float v8f __attribute__((ext_vector_type(8)));

// ---------------------------------------------------------------------------
// kNN on raw 3-D coords: one 256-thread block per (b, n). neg squared dist in
// LDS, then 40 argmax passes (tie -> lower index, matching jax top_k).
// ---------------------------------------------------------------------------
__global__ __launch_bounds__(256) void knn3_kernel(const float* __restrict__ x,
                                                   int* __restrict__ idx) {
  int b = blockIdx.y, n = blockIdx.x, t = threadIdx.x;
  __shared__ float dist[NPTS];
  __shared__ float redv[256];
  __shared__ int   redi[256];
  const float* xb = x + (size_t)b * NPTS * 3;
  float px = xb[(size_t)n * 3 + 0], py = xb[(size_t)n * 3 + 1], pz = xb[(size_t)n * 3 + 2];
  for (int m = t; m < NPTS; m += 256) {
    float dx = xb[(size_t)m * 3 + 0] - px;
    float dy = xb[(size_t)m * 3 + 1] - py;
    float dz = xb[(size_t)m * 3 + 2] - pz;
    dist[m] = -(dx * dx + dy * dy + dz * dz);
  }
  __syncthreads();
  int* out = idx + ((size_t)b * NPTS + n) * KNBR;
  for (int kk = 0; kk < KNBR; ++kk) {
    float best = -3.4e38f; int bi = NPTS;
    for (int m = t; m < NPTS; m += 256) {
      float v = dist[m];
      if (v > best) { best = v; bi = m; }
    }
    redv[t] = best; redi[t] = bi;
    __syncthreads();
    for (int s = 128; s > 0; s >>= 1) {
      if (t < s) {
        float ov = redv[t + s]; int oi = redi[t + s];
        if (ov > redv[t] || (ov == redv[t] && oi < redi[t])) { redv[t] = ov; redi[t] = oi; }
      }
      __syncthreads();
    }
    if (t == 0) { out[kk] = redi[0]; dist[redi[0]] = -3.4e38f; }
    __syncthreads();
  }
}

// ---------------------------------------------------------------------------
// kNN on 64-channel features stored channel-major in h[b][c][n].
// ---------------------------------------------------------------------------
__global__ __launch_bounds__(256) void knn64_kernel(const float* __restrict__ h,
                                                    int* __restrict__ idx) {
  int b = blockIdx.y, n = blockIdx.x, t = threadIdx.x;
  __shared__ float q[64];
  __shared__ float dist[NPTS];
  __shared__ float redv[256];
  __shared__ int   redi[256];
  const float* hb = h + (size_t)b * 128 * NPTS;
  if (t < 64) q[t] = hb[(size_t)t * NPTS + n];
  __syncthreads();
  for (int m = t; m < NPTS; m += 256) {
    float s = 0.f;
    for (int c = 0; c < 64; ++c) {
      float d = hb[(size_t)c * NPTS + m] - q[c];
      s += d * d;
    }
    dist[m] = -s;
  }
  __syncthreads();
  int* out = idx + ((size_t)b * NPTS + n) * KNBR;
  for (int kk = 0; kk < KNBR; ++kk) {
    float best = -3.4e38f; int bi = NPTS;
    for (int m = t; m < NPTS; m += 256) {
      float v = dist[m];
      if (v > best) { best = v; bi = m; }
    }
    redv[t] = best; redi[t] = bi;
    __syncthreads();
    for (int s = 128; s > 0; s >>= 1) {
      if (t < s) {
        float ov = redv[t + s]; int oi = redi[t + s];
        if (ov > redv[t] || (ov == redv[t] && oi < redi[t])) { redv[t] = ov; redi[t] = oi; }
      }
      __syncthreads();
    }
    if (t == 0) { out[kk] = redi[0]; dist[redi[0]] = -3.4e38f; }
    __syncthreads();
  }
}

// ---------------------------------------------------------------------------
// Edge-conv stage 1: feat = [nbr-ctr(3), ctr(3)] -> conv 6->64 (bn,lrelu)
// -> conv 64->64 (bn,lrelu) -> max over k.  One 64-thread block per (b,n);
// thread t owns output channel t.  Writes h[b][t][n] (channels 0..63).
// ---------------------------------------------------------------------------
__global__ __launch_bounds__(64) void edgeconv1_kernel(
    const float* __restrict__ x, const int* __restrict__ idx,
    const float* __restrict__ w1, const float* __restrict__ g1, const float* __restrict__ b1,
    const float* __restrict__ w2, const float* __restrict__ g2, const float* __restrict__ b2,
    float* __restrict__ h) {
  int b = blockIdx.y, n = blockIdx.x, t = threadIdx.x;
  __shared__ float nbr[KNBR][3];
  __shared__ float ctr[3];
  __shared__ float y1[64][KNBR + 1];
  __shared__ float w2s[64][65];
  const float* xb = x + (size_t)b * NPTS * 3;
  const int* id = idx + ((size_t)b * NPTS + n) * KNBR;
  if (t < 3) ctr[t] = xb[(size_t)n * 3 + t];
  for (int e = t; e < KNBR * 3; e += 64) {
    int k = e / 3, c = e % 3;
    nbr[k][c] = xb[(size_t)id[k] * 3 + c];
  }
  for (int e = t; e < 64 * 64; e += 64) w2s[e >> 6][e & 63] = w2[e];
  __syncthreads();
  float wr[6];
  #pragma unroll
  for (int i = 0; i < 6; ++i) wr[i] = w1[t * 6 + i];
  float s1 = g1[t] * BN_RSQ, o1 = b1[t];
  float base = wr[3] * ctr[0] + wr[4] * ctr[1] + wr[5] * ctr[2];
  for (int k = 0; k < KNBR; ++k) {
    float v = base + wr[0] * (nbr[k][0] - ctr[0]) + wr[1] * (nbr[k][1] - ctr[1])
                   + wr[2] * (nbr[k][2] - ctr[2]);
    v = v * s1 + o1;
    y1[t][k] = (v > 0.f) ? v : 0.2f * v;
  }
  __syncthreads();
  float s2 = g2[t] * BN_RSQ, o2 = b2[t];
  float mx = -3.4e38f;
  for (int k = 0; k < KNBR; ++k) {
    float v = 0.f;
    for (int c = 0; c < 64; ++c) v += w2s[t][c] * y1[c][k];
    v = v * s2 + o2;
    v = (v > 0.f) ? v : 0.2f * v;
    mx = fmaxf(mx, v);
  }
  h[((size_t)b * 128 + t) * NPTS + n] = mx;
}

// ---------------------------------------------------------------------------
// Edge-conv stage 2: input = h channels 0..63, feat = [nbr-ctr(64), ctr(64)],
// conv 128->64, conv 64->64, max over k. Writes h channels 64..127.
// ---------------------------------------------------------------------------
__global__ __launch_bounds__(64) void edgeconv2_kernel(
    const float* __restrict__ h, const int* __restrict__ idx,
    const float* __restrict__ w3, const float* __restrict__ g3, const float* __restrict__ b3,
    const float* __restrict__ w4, const float* __restrict__ g4, const float* __restrict__ b4,
    float* __restrict__ hout) {
  int b = blockIdx.y, n = blockIdx.x, t = threadIdx.x;
  __shared__ float nbrf[KNBR][64];
  __shared__ float ctrf[64];
  __shared__ float y3[64][KNBR + 1];
  __shared__ float w3s[64][129];
  const float* hb = h + (size_t)b * 128 * NPTS;
  const int* id = idx + ((size_t)b * NPTS + n) * KNBR;
  ctrf[t] = hb[(size_t)t * NPTS + n];
  for (int e = t; e < KNBR * 64; e += 64) {
    int k = e >> 6, c = e & 63;
    nbrf[k][c] = hb[(size_t)c * NPTS + id[k]];
  }
  for (int e = t; e < 64 * 128; e += 64) w3s[e >> 7][e & 127] = w3[e];
  __syncthreads();
  float s3 = g3[t] * BN_RSQ, o3 = b3[t];
  float base = 0.f;
  for (int c = 0; c < 64; ++c) base += w3s[t][64 + c] * ctrf[c];
  for (int k = 0; k < KNBR; ++k) {
    float v = base;
    for (int c = 0; c < 64; ++c) v += w3s[t][c] * (nbrf[k][c] - ctrf[c]);
    v = v * s3 + o3;
    y3[t][k] = (v > 0.f) ? v : 0.2f * v;
  }
  __syncthreads();
  float s4 = g4[t] * BN_RSQ, o4 = b4[t];
  const float* w4r = w4 + (size_t)t * 64;  // small, L2-resident
  float mx = -3.4e38f;
  for (int k = 0; k < KNBR; ++k) {
    float v = 0.f;
    for (int c = 0; c < 64; ++c) v += w4r[c] * y3[c][k];
    v = v * s4 + o4;
    v = (v > 0.f) ? v : 0.2f * v;
    mx = fmaxf(mx, v);
  }
  hout[((size_t)b * 128 + 64 + t) * NPTS + n] = mx;
}

// ---------------------------------------------------------------------------
// Generic fp32 WMMA GEMM: Out[b][m][n] = sum_k W[m][k] * X[b][k][n].
// One wave per 16x16 output tile, K stepped by 4 via V_WMMA_F32_16X16X4_F32.
// A layout: lane L -> M=L%16; vgpr0 holds K=k0+2*half, vgpr1 K=k0+2*half+1.
// ---------------------------------------------------------------------------
__global__ __launch_bounds__(32) void gemm_wmma_f32(
    const float* __restrict__ W, const float* __restrict__ X,
    float* __restrict__ Out, int M, int K, int N) {
  int nt = blockIdx.x, mt = blockIdx.y, b = blockIdx.z;
  int lane = threadIdx.x;
  int half = lane >> 4;
  int l16  = lane & 15;
  const float* Wm = W + (size_t)(mt * 16 + l16) * K;
  const float* Xb = X + (size_t)b * K * N;
  v8f acc = {};
  for (int k0 = 0; k0 < K; k0 += 4) {
    v2f a, bv;
    int ka = k0 + 2 * half;
    a.x = Wm[ka];
    a.y = Wm[ka + 1];
    bv.x = Xb[(size_t)ka * N + nt * 16 + l16];
    bv.y = Xb[(size_t)(ka + 1) * N + nt * 16 + l16];
    acc = __builtin_amdgcn_wmma_f32_16x16x4_f32(false, a, false, bv,
                                                (short)0, acc, false, false);
  }
  float* Ob = Out + (size_t)b * M * N;
  #pragma unroll
  for (int r = 0; r < 8; ++r) {
    int m = mt * 16 + r + 8 * half;   // C layout: vgpr r -> M=r (lo half) / r+8
    Ob[(size_t)m * N + nt * 16 + l16] = acc[r];
  }
}

// ---------------------------------------------------------------------------
// Attention pass 1: per (b, n) compute rowmax/rowsum of energy[n, :] where
// energy[n,m] = dot32(qkx[:,n], qkx[:,m]).
// ---------------------------------------------------------------------------
__global__ __launch_bounds__(256) void attn_rowstats_kernel(
    const float* __restrict__ qkx, float* __restrict__ rowmax, float* __restrict__ rowsum) {
  int b = blockIdx.y, n = blockIdx.x, t = threadIdx.x;
  __shared__ float q[32];
  __shared__ float ebuf[NPTS];
  __shared__ float red[256];
  const float* Q = qkx + (size_t)b * 32 * NPTS;
  if (t < 32) q[t] = Q[(size_t)t * NPTS + n];
  __syncthreads();
  for (int m = t; m < NPTS; m += 256) {
    float s = 0.f;
    for (int c = 0; c < 32; ++c) s += q[c] * Q[(size_t)c * NPTS + m];
    ebuf[m] = s;
  }
  __syncthreads();
  float lm = -3.4e38f;
  for (int m = t; m < NPTS; m += 256) lm = fmaxf(lm, ebuf[m]);
  red[t] = lm; __syncthreads();
  for (int s = 128; s > 0; s >>= 1) { if (t < s) red[t] = fmaxf(red[t], red[t + s]); __syncthreads(); }
  float rm = red[0]; __syncthreads();
  float ls = 0.f;
  for (int m = t; m < NPTS; m += 256) ls += __expf(ebuf[m] - rm);
  red[t] = ls; __syncthreads();
  for (int s = 128; s > 0; s >>= 1) { if (t < s) red[t] += red[t + s]; __syncthreads(); }
  if (t == 0) { rowmax[(size_t)b * NPTS + n] = rm; rowsum[(size_t)b * NPTS + n] = red[0]; }
}

// ---------------------------------------------------------------------------
// Attention pass 2 (flash-style): per (b, 16-column tile m0) stream n-chunks
// of 16, build P[nn][m]=softmax tile in LDS, accumulate xv·P with WMMA
// (8 waves cover the 128 channels) and the column sums; finish with the
// column renorm and d = h - xr.
// ---------------------------------------------------------------------------
__global__ __launch_bounds__(256) void attn_apply_kernel(
    const float* __restrict__ qkx, const float* __restrict__ xv,
    const float* __restrict__ hbuf, const float* __restrict__ rowmax,
    const float* __restrict__ rowsum, float* __restrict__ dbuf) {
  int b = blockIdx.y;
  int m0 = blockIdx.x * 16;
  int tid = threadIdx.x;
  int lane = tid & 31, wave = tid >> 5;
  int half = (lane >> 4) & 1, l16 = lane & 15;
  __shared__ float P[16][17];
  __shared__ float Qc[32][16];
  __shared__ float Qr[32][17];
  __shared__ float csum_sh[16][17];
  __shared__ float colsum[16];
  const float* Q = qkx + (size_t)b * 32 * NPTS;
  const float* V = xv + (size_t)b * 128 * NPTS;
  const float* rmaxb = rowmax + (size_t)b * NPTS;
  const float* rsumb = rowsum + (size_t)b * NPTS;
  for (int e = tid; e < 32 * 16; e += 256) Qc[e >> 4][e & 15] = Q[(size_t)(e >> 4) * NPTS + m0 + (e & 15)];
  int prow = tid >> 4;   // n within chunk
  int pcol = tid & 15;   // m within tile
  float csum_loc = 0.f;
  v8f acc = {};
  __syncthreads();
  for (int n0 = 0; n0 < NPTS; n0 += 16) {
    for (int e = tid; e < 32 * 16; e += 256) Qr[e >> 4][e & 15] = Q[(size_t)(e >> 4) * NPTS + n0 + (e & 15)];
    __syncthreads();
    float s = 0.f;
    for (int c = 0; c < 32; ++c) s += Qr[c][prow] * Qc[c][pcol];
    int n = n0 + prow;
    float p = __expf(s - rmaxb[n]) / rsumb[n];
    P[prow][pcol] = p;
    csum_loc += p;
    __syncthreads();
    int ch = wave * 16 + l16;   // A-matrix row (xv channel)
    #pragma unroll
    for (int ks = 0; ks < 4; ++ks) {
      v2f a, bv;
      int kk = n0 + ks * 4 + 2 * half;
      a.x = V[(size_t)ch * NPTS + kk];
      a.y = V[(size_t)ch * NPTS + kk + 1];
      bv.x = P[ks * 4 + 2 * half][l16];
      bv.y = P[ks * 4 + 2 * half + 1][l16];
      acc = __builtin_amdgcn_wmma_f32_16x16x4_f32(false, a, false, bv,
                                                  (short)0, acc, false, false);
    }
    __syncthreads();
  }
  csum_sh[prow][pcol] = csum_loc;
  __syncthreads();
  if (tid < 16) {
    float s = 0.f;
    for (int r = 0; r < 16; ++r) s += csum_sh[r][tid];
    colsum[tid] = s + 1e-9f;
  }
  __syncthreads();
  const float* H = hbuf + (size_t)b * 128 * NPTS;
  float* D = dbuf + (size_t)b * 128 * NPTS;
  #pragma unroll
  for (int r = 0; r < 8; ++r) {
    int c = wave * 16 + r + 8 * half;
    int m = m0 + l16;
    float xr = acc[r] / colsum[l16];
    D[(size_t)c * NPTS + m] = H[(size_t)c * NPTS + m] - xr;
  }
}

// ---------------------------------------------------------------------------
// Elementwise helpers
// ---------------------------------------------------------------------------
__global__ __launch_bounds__(256) void add_bias_kernel(
    float* __restrict__ y, const float* __restrict__ bias, int C, int total) {
  int i = blockIdx.x * blockDim.x + threadIdx.x;
  if (i >= total) return;
  int c = (i / NPTS) % C;
  y[i] += bias[c];
}

__global__ __launch_bounds__(256) void sa_update_kernel(
    float* __restrict__ h, const float* __restrict__ dr,
    const float* __restrict__ tb, const float* __restrict__ g,
    const float* __restrict__ bb, float* __restrict__ cat, int layer_off, int total) {
  int i = blockIdx.x * blockDim.x + threadIdx.x;
  if (i >= total) return;
  int n = i % NPTS;
  int c = (i / NPTS) & 127;
  int b = i / (NPTS * 128);
  float sc = g[c] * BN_RSQ;
  float v = (dr[i] + tb[c]) * sc + bb[c];
  float hn = h[i] + fmaxf(v, 0.f);
  h[i] = hn;
  cat[((size_t)b * 512 + layer_off + c) * NPTS + n] = hn;
}

__global__ __launch_bounds__(256) void bn_lrelu_kernel(
    float* __restrict__ y, const float* __restrict__ g,
    const float* __restrict__ bb, int C, int total) {
  int i = blockIdx.x * blockDim.x + threadIdx.x;
  if (i >= total) return;
  int c = (i / NPTS) % C;
  float v = y[i] * (g[c] * BN_RSQ) + bb[c];
  y[i] = (v > 0.f) ? v : 0.2f * v;
}

// ---------------------------------------------------------------------------
extern "C" void kernel_launch(void* const* d_in, const int* in_sizes, int n_in,
                              void* d_out, int out_size, void* d_ws, size_t ws_size,
                              hipStream_t stream) {
  const float* x  = (const float*)d_in[0];
  const float* w1 = (const float*)d_in[1];  const float* g1 = (const float*)d_in[2];  const float* b1 = (const float*)d_in[3];
  const float* w2 = (const float*)d_in[4];  const float* g2 = (const float*)d_in[5];  const float* b2 = (const float*)d_in[6];
  const float* w3 = (const float*)d_in[7];  const float* g3 = (const float*)d_in[8];  const float* b3 = (const float*)d_in[9];
  const float* w4 = (const float*)d_in[10]; const float* g4 = (const float*)d_in[11]; const float* b4 = (const float*)d_in[12];
  // d_in[13..15] = w5,g5,b5 (unused by the reference)
  const float* sa_qk = (const float*)d_in[16];
  const float* sa_v  = (const float*)d_in[17];
  const float* sa_vb = (const float*)d_in[18];
  const float* sa_t  = (const float*)d_in[19];
  const float* sa_tb = (const float*)d_in[20];
  const float* sa_g  = (const float*)d_in[21];
  const float* sa_b  = (const float*)d_in[22];
  const float* fuse_w = (const float*)d_in[23];
  const float* fuse_g = (const float*)d_in[24];
  const float* fuse_b = (const float*)d_in[25];
  float* out = (float*)d_out;
  const int B = in_sizes[0] / (NPTS * 3);

  float* wsf = (float*)d_ws;
  size_t o = 0;
  float* h     = wsf + o; o += (size_t)B * 128 * NPTS;
  float* cat   = wsf + o; o += (size_t)B * 512 * NPTS;
  float* qkx   = wsf + o; o += (size_t)B * 32  * NPTS;
  float* xv    = wsf + o; o += (size_t)B * 128 * NPTS;
  float* dbuf  = wsf + o; o += (size_t)B * 128 * NPTS;
  float* drbuf = wsf + o; o += (size_t)B * 128 * NPTS;
  float* rmax  = wsf + o; o += (size_t)B * NPTS;
  float* rsum  = wsf + o; o += (size_t)B * NPTS;
  int* idx1 = (int*)(wsf + o); o += (size_t)B * NPTS * KNBR;
  int* idx2 = (int*)(wsf + o);

  dim3 bpn(NPTS, B);
  knn3_kernel<<<bpn, 256, 0, stream>>>(x, idx1);
  edgeconv1_kernel<<<bpn, 64, 0, stream>>>(x, idx1, w1, g1, b1, w2, g2, b2, h);
  knn64_kernel<<<bpn, 256, 0, stream>>>(h, idx2);
  edgeconv2_kernel<<<bpn, 64, 0, stream>>>(h, idx2, w3, g3, b3, w4, g4, b4, h);

  const int totalH = B * 128 * NPTS;
  const int egrd = (totalH + 255) / 256;
  for (int L = 0; L < 4; ++L) {
    const float* qk = sa_qk + (size_t)L * 32 * 128;
    const float* vw = sa_v  + (size_t)L * 128 * 128;
    const float* vb = sa_vb + (size_t)L * 128;
    const float* tw = sa_t  + (size_t)L * 128 * 128;
    const float* tb = sa_tb + (size_t)L * 128;
    const float* gg = sa_g  + (size_t)L * 128;
    const float* bb = sa_b  + (size_t)L * 128;
    gemm_wmma_f32<<<dim3(NPTS / 16, 2, B),  32, 0, stream>>>(qk, h, qkx, 32, 128, NPTS);
    gemm_wmma_f32<<<dim3(NPTS / 16, 8, B),  32, 0, stream>>>(vw, h, xv, 128, 128, NPTS);
    add_bias_kernel<<<egrd, 256, 0, stream>>>(xv, vb, 128, totalH);
    attn_rowstats_kernel<<<bpn, 256, 0, stream>>>(qkx, rmax, rsum);
    attn_apply_kernel<<<dim3(NPTS / 16, B), 256, 0, stream>>>(qkx, xv, h, rmax, rsum, dbuf);
    gemm_wmma_f32<<<dim3(NPTS / 16, 8, B),  32, 0, stream>>>(tw, dbuf, drbuf, 128, 128, NPTS);
    sa_update_kernel<<<egrd, 256, 0, stream>>>(h, drbuf, tb, gg, bb, cat, L * 128, totalH);
  }
  gemm_wmma_f32<<<dim3(NPTS / 16, 16, B), 32, 0, stream>>>(fuse_w, cat, out, 256, 512, NPTS);
  const int totalO = B * 256 * NPTS;
  bn_lrelu_kernel<<<(totalO + 255) / 256, 256, 0, stream>>>(out, fuse_g, fuse_b, 256, totalO);

  (void)n_in; (void)out_size; (void)ws_size;
}